// WeightedHashEmbedding_30623116820708
// MI455X (gfx1250) — compile-verified
//
#include <hip/hip_runtime.h>
#include <stdint.h>

#define TDIM      64
#define NCHUNK    32
#define WAVES_PB  8          // 8 waves (batches) per 256-thread block

// out[b,:] = sqrt(2) * sum_c w[b,c] * table[h0[b,c], :]
//   w[b,c] = table_flat[h1[b,c]]
__global__ __launch_bounds__(256) void whe_kernel(
    const float* __restrict__ table,
    const int*   __restrict__ h0,
    const int*   __restrict__ h1,
    float*       __restrict__ out,
    int nbatch)
{
    __shared__ int sh_h0[256];
    __shared__ int sh_h1[256];

    const int t    = threadIdx.x;
    const int wave = t >> 5;
    const int lane = t & 31;
    const int b    = blockIdx.x * WAVES_PB + wave;   // one batch per wave

    if (b >= nbatch) return;                          // wave-uniform guard

    // ---- CDNA5 async global->LDS staging of this wave's 32+32 indices.
    //      Thread t stages element (b*32 + lane) of each index array, i.e.
    //      each wave stages exactly its own batch -> no barrier required,
    //      just s_wait_asynccnt. Tracked by ASYNCcnt (gfx1250 path).
    {
        const int gidx = blockIdx.x * 256 + t;        // == b*NCHUNK + lane
        uint32_t lds0 = (uint32_t)(uintptr_t)(&sh_h0[t]);
        uint32_t lds1 = (uint32_t)(uintptr_t)(&sh_h1[t]);
        const int* g0 = h0 + gidx;
        const int* g1 = h1 + gidx;
        asm volatile(
            "global_load_async_to_lds_b32 %0, %2, off\n\t"
            "global_load_async_to_lds_b32 %1, %3, off\n\t"
            "s_wait_asynccnt 0"
            :
            : "v"(lds0), "v"(lds1), "v"(g0), "v"(g1)
            : "memory");
    }

    // Lane l now reads h0[b,l] / h1[b,l] from LDS.
    const int my_h0 = sh_h0[t];
    const int my_h1 = sh_h1[t];

    // Per-lane scalar weight gather: w[b,lane] = table_flat[h1[b,lane]]
    const float wv = table[(size_t)(unsigned)my_h1];

    // Accumulate sqrt(2) * sum_c w_c * row_c ; lane covers dims {2*lane, 2*lane+1}
    float accx = 0.0f, accy = 0.0f;

#pragma unroll
    for (int c = 0; c < NCHUNK; ++c) {
        // Broadcast chunk c's row index and weight to SGPRs (wave-uniform):
        // enables SADDR-form global_load_b64 and scalar-operand v_fmac.
        int   r  = __builtin_amdgcn_readlane(my_h0, c);
        int   wi = __builtin_amdgcn_readlane(__builtin_bit_cast(int, wv), c);
        float w  = __builtin_bit_cast(float, wi);

        // Uniform row base + lane*8B -> one coalesced 256B request per wave.
        const float2* row = (const float2*)(table + (size_t)(unsigned)r * TDIM);
        float2 v = row[lane];

        accx = fmaf(w, v.x, accx);
        accy = fmaf(w, v.y, accy);
    }

    const float scale = 1.41421356237309515f;         // sqrt(2048)/32 = sqrt(2)
    float2 res;
    res.x = accx * scale;
    res.y = accy * scale;

    // Coalesced 256B store per wave.
    ((float2*)(out + (size_t)b * TDIM))[lane] = res;
}

extern "C" void kernel_launch(void* const* d_in, const int* in_sizes, int n_in,
                              void* d_out, int out_size, void* d_ws, size_t ws_size,
                              hipStream_t stream) {
    const float* table = (const float*)d_in[0];   // [ROWS, 64] f32
    const int*   h0    = (const int*)  d_in[1];   // [B, 32] i32 row indices
    const int*   h1    = (const int*)  d_in[2];   // [B, 32] i32 flat indices
    float*       out   = (float*)d_out;           // [B, 64] f32

    const int nbatch = in_sizes[1] / NCHUNK;      // 16384
    const int blocks = (nbatch + WAVES_PB - 1) / WAVES_PB;

    whe_kernel<<<blocks, 256, 0, stream>>>(table, h0, h1, out, nbatch);
}